// StructuralAttentionLayer_12575664243410
// MI455X (gfx1250) — compile-verified
//
#include <hip/hip_runtime.h>

#define N_NODES 50000
#define N_EDGES 1600000
#define IN_DIM  128
#define N_HEADS 8
#define OUT_DIM 16
#define HC      128   // N_HEADS * OUT_DIM

typedef __attribute__((ext_vector_type(2))) float v2f;
typedef __attribute__((ext_vector_type(8))) float v8f;

// monotonic float<->uint mapping for atomicMax on floats
__device__ __forceinline__ unsigned f2ord(float f) {
    unsigned u = __float_as_uint(f);
    return (u & 0x80000000u) ? ~u : (u | 0x80000000u);
}
__device__ __forceinline__ float ord2f(unsigned u) {
    return (u & 0x80000000u) ? __uint_as_float(u ^ 0x80000000u)
                             : __uint_as_float(~u);
}

// ---------------------------------------------------------------------------
// K0: initialize segment-max (encoded -inf), denom, accumulator
// ---------------------------------------------------------------------------
__global__ void gat_init(unsigned* __restrict__ m_ord,
                         float* __restrict__ denom,
                         float* __restrict__ acc) {
    int i = blockIdx.x * blockDim.x + threadIdx.x;
    if (i < N_NODES * N_HEADS) {
        m_ord[i] = 0x007FFFFFu;   // f2ord(-inf)
        denom[i] = 0.0f;
    }
    if (i < N_NODES * HC) acc[i] = 0.0f;
}

// ---------------------------------------------------------------------------
// K1: x = feature @ w_lin ; res(d_out) = feature @ w_res ; al/ar reductions
// One wave per 16x16 output tile. n_tile 0..7 -> w_lin (head == n_tile),
// n_tile 8..15 -> w_res. 50000 % 16 == 0, so no M guard needed.
// ---------------------------------------------------------------------------
__global__ void gat_gemm_wmma(const float* __restrict__ feat,
                              const float* __restrict__ w_lin,
                              const float* __restrict__ w_res,
                              const float* __restrict__ att_l,
                              const float* __restrict__ att_r,
                              float* __restrict__ x,
                              float* __restrict__ res_out,
                              float* __restrict__ al,
                              float* __restrict__ ar) {
    const int wave   = threadIdx.x >> 5;
    const int lane   = threadIdx.x & 31;
    const int tile   = blockIdx.x * 4 + wave;   // 50000 tiles total
    const int m_tile = tile >> 4;               // 0..3124
    const int n_tile = tile & 15;               // 0..15
    const int half   = lane >> 4;               // 0 or 1
    const int l16    = lane & 15;

    const bool is_lin = (n_tile < 8);
    const float* __restrict__ W = is_lin ? w_lin : w_res;
    const int ncol0 = (n_tile & 7) * 16;

    const float* __restrict__ Arow = feat + (size_t)(m_tile * 16 + l16) * IN_DIM;

    v8f c = {};
#pragma unroll 8
    for (int k0 = 0; k0 < IN_DIM; k0 += 4) {
        // A 16x4: lanes 0-15 hold K={k0,k0+1}, lanes 16-31 hold K={k0+2,k0+3}
        const int ka = k0 + half * 2;
        float2 av = *(const float2*)(Arow + ka);
        v2f a; a.x = av.x; a.y = av.y;
        // B 4x16: VGPR0 = row ka, VGPR1 = row ka+1 (mirrored half-wave split)
        v2f b;
        b.x = W[(size_t)ka * HC + ncol0 + l16];
        b.y = W[(size_t)(ka + 1) * HC + ncol0 + l16];
        c = __builtin_amdgcn_wmma_f32_16x16x4_f32(
                /*neg_a=*/false, a, /*neg_b=*/false, b,
                /*c_mod=*/(short)0, c, /*reuse_a=*/false, /*reuse_b=*/false);
    }

    // C/D 16x16: VGPR v -> rows v (lanes 0-15) and v+8 (lanes 16-31)
    float* __restrict__ dst = is_lin ? x : res_out;
#pragma unroll
    for (int v = 0; v < 8; ++v) {
        int row = m_tile * 16 + v + half * 8;
        dst[(size_t)row * HC + ncol0 + l16] = c[v];
    }

    if (is_lin) {
        // head h == n_tile; reduce al/ar for the 16 rows of this tile
        const int h = n_tile;
        const float wl = att_l[h * OUT_DIM + l16];
        const float wr = att_r[h * OUT_DIM + l16];
#pragma unroll
        for (int v = 0; v < 8; ++v) {
            float sl = c[v] * wl;
            float sr = c[v] * wr;
#pragma unroll
            for (int off = 1; off < 16; off <<= 1) {
                sl += __shfl_xor(sl, off, 32);
                sr += __shfl_xor(sr, off, 32);
            }
            if (l16 == 0) {
                int row = m_tile * 16 + v + half * 8;
                al[row * N_HEADS + h] = sl;
                ar[row * N_HEADS + h] = sr;
            }
        }
    }
}

// ---------------------------------------------------------------------------
// K2: per-edge logits, leaky relu, atomic segment-max (ordered-uint)
// ---------------------------------------------------------------------------
__global__ void gat_edge_logits(const int* __restrict__ esrc,
                                const int* __restrict__ edst,
                                const float* __restrict__ ew,
                                const float* __restrict__ al,
                                const float* __restrict__ ar,
                                float* __restrict__ alpha,
                                unsigned* __restrict__ m_ord) {
    int e = blockIdx.x * blockDim.x + threadIdx.x;
    if (e >= N_EDGES) return;
    int s = esrc[e], d = edst[e];
    float w = ew[e];
    const float4* alp = (const float4*)(al + (size_t)s * N_HEADS);
    const float4* arp = (const float4*)(ar + (size_t)d * N_HEADS);
    float4 a0 = alp[0], a1 = alp[1];
    float4 b0 = arp[0], b1 = arp[1];
    float v[8] = {a0.x + b0.x, a0.y + b0.y, a0.z + b0.z, a0.w + b0.w,
                  a1.x + b1.x, a1.y + b1.y, a1.z + b1.z, a1.w + b1.w};
#pragma unroll
    for (int h = 0; h < 8; ++h) {
        float t = w * v[h];
        t = (t > 0.0f) ? t : 0.2f * t;          // leaky_relu 0.2
        alpha[(size_t)e * N_HEADS + h] = t;
        atomicMax(m_ord + (size_t)d * N_HEADS + h, f2ord(t));
    }
}

// ---------------------------------------------------------------------------
// K3: e = exp(alpha - m[dst]); denom += e  (alpha overwritten in place)
// ---------------------------------------------------------------------------
__global__ void gat_edge_exp(const int* __restrict__ edst,
                             const unsigned* __restrict__ m_ord,
                             float* __restrict__ alpha,
                             float* __restrict__ denom) {
    int e = blockIdx.x * blockDim.x + threadIdx.x;
    if (e >= N_EDGES) return;
    int d = edst[e];
#pragma unroll
    for (int h = 0; h < 8; ++h) {
        float m = ord2f(m_ord[(size_t)d * N_HEADS + h]);
        if (m < -1e38f) m = 0.0f;               // isfinite guard (only -inf possible)
        float ex = __expf(alpha[(size_t)e * N_HEADS + h] - m);
        alpha[(size_t)e * N_HEADS + h] = ex;
        atomicAdd(denom + (size_t)d * N_HEADS + h, ex);
    }
}

// ---------------------------------------------------------------------------
// K4: wave-per-edge message scatter: acc[dst] += coef * x[src]
// lane handles 4 consecutive channels; head = lane>>2
// ---------------------------------------------------------------------------
__global__ void gat_scatter(const int* __restrict__ esrc,
                            const int* __restrict__ edst,
                            const float* __restrict__ eArr,
                            const float* __restrict__ denom,
                            const float* __restrict__ x,
                            float* __restrict__ acc) {
    int gid = blockIdx.x * blockDim.x + threadIdx.x;
    int e = gid >> 5;
    if (e >= N_EDGES) return;
    int lane = gid & 31;
    int c0 = lane * 4;
    int h  = lane >> 2;
    int s = esrc[e], d = edst[e];
    float coef = eArr[(size_t)e * N_HEADS + h] / denom[(size_t)d * N_HEADS + h];
    float4 xv = *(const float4*)(x + (size_t)s * HC + c0);
    float* ap = acc + (size_t)d * HC + c0;
    atomicAdd(ap + 0, xv.x * coef);
    atomicAdd(ap + 1, xv.y * coef);
    atomicAdd(ap + 2, xv.z * coef);
    atomicAdd(ap + 3, xv.w * coef);
}

// ---------------------------------------------------------------------------
// K5: out = elu(acc) + residual (residual already resides in d_out)
// ---------------------------------------------------------------------------
__global__ void gat_final(const float* __restrict__ acc,
                          float* __restrict__ out) {
    int i = blockIdx.x * blockDim.x + threadIdx.x;
    if (i >= N_NODES * HC) return;
    float a = acc[i];
    float eluv = (a > 0.0f) ? a : (__expf(a) - 1.0f);
    out[i] = eluv + out[i];
}

// ---------------------------------------------------------------------------
extern "C" void kernel_launch(void* const* d_in, const int* in_sizes, int n_in,
                              void* d_out, int out_size, void* d_ws, size_t ws_size,
                              hipStream_t stream) {
    const float* feat  = (const float*)d_in[0];
    const int*   esrc  = (const int*)d_in[1];
    const int*   edst  = (const int*)d_in[2];
    const float* ew    = (const float*)d_in[3];
    const float* w_lin = (const float*)d_in[4];
    const float* att_l = (const float*)d_in[5];
    const float* att_r = (const float*)d_in[6];
    const float* w_res = (const float*)d_in[7];
    float* out = (float*)d_out;

    // workspace layout (floats)
    float* ws    = (float*)d_ws;
    float* x     = ws;                                        // N*128
    float* al    = x + (size_t)N_NODES * HC;                  // N*8
    float* ar    = al + (size_t)N_NODES * N_HEADS;            // N*8
    unsigned* m_ord = (unsigned*)(ar + (size_t)N_NODES * N_HEADS); // N*8
    float* denom = (float*)(m_ord + (size_t)N_NODES * N_HEADS);    // N*8
    float* alpha = denom + (size_t)N_NODES * N_HEADS;         // E*8 (reused as e)
    float* acc   = alpha + (size_t)N_EDGES * N_HEADS;         // N*128

    const int initN = N_NODES * HC;
    gat_init<<<(initN + 255) / 256, 256, 0, stream>>>(m_ord, denom, acc);

    // 3125 M-tiles * 16 N-tiles = 50000 wave-tiles; 4 waves / 128-thread block
    gat_gemm_wmma<<<12500, 128, 0, stream>>>(feat, w_lin, w_res, att_l, att_r,
                                             x, out, al, ar);

    gat_edge_logits<<<(N_EDGES + 255) / 256, 256, 0, stream>>>(
        esrc, edst, ew, al, ar, alpha, m_ord);

    gat_edge_exp<<<(N_EDGES + 255) / 256, 256, 0, stream>>>(
        edst, m_ord, alpha, denom);

    gat_scatter<<<(N_EDGES * 32 + 255) / 256, 256, 0, stream>>>(
        esrc, edst, alpha, denom, x, acc);

    gat_final<<<(initN + 255) / 256, 256, 0, stream>>>(acc, out);
}